// SliceAttention_43173011259490
// MI455X (gfx1250) — compile-verified
//
#include <hip/hip_runtime.h>

typedef __attribute__((ext_vector_type(2))) float v2f;
typedef __attribute__((ext_vector_type(8))) float v8f;

#define B_ 4
#define N_ 16384
#define HID_ 512
#define H_ 8
#define M_ 32
#define D_ 64

// d_out region offsets (floats): out, slice_weights, alpha, zeros(H,M), attn
#define SW_OFF    (B_*N_*HID_)              // 33554432
#define ALPHA_OFF (SW_OFF + B_*H_*M_*N_)    // 50331648
#define ZEROS_OFF (ALPHA_OFF + H_)          // 50331656
#define ATTN_OFF  (ZEROS_OFF + H_*M_)       // 50331912

// workspace offsets (floats)
#define WS_ST  0                            // raw slice_token accum  B*H*M*D
#define WS_RS  (B_*H_*M_*D_)                // rowsums  B*H*M
#define WS_STN (WS_RS + B_*H_*M_)           // normalized slice_token (B,M,HID)
#define WS_WT  (WS_STN + B_*M_*HID_)        // transposed wkv head slices (H,128,512)
#define WS_GT  (WS_WT + H_*128*HID_)        // transposed folded G (B,512,256)

// padded LDS row stride (even -> b64-aligned; (2*lane+k)%64 spreads banks)
#define LP 66

__device__ __forceinline__ v8f wmma4(v2f a, v2f b, v8f c) {
  // V_WMMA_F32_16X16X4_F32: (neg_a, A, neg_b, B, c_mod, C, reuse_a, reuse_b)
  return __builtin_amdgcn_wmma_f32_16x16x4_f32(false, a, false, b, (short)0, c,
                                               false, false);
}

// ---------------------------------------------------------------------------
// init: zero atomic accumulators; emit alpha and zeros outputs
// ---------------------------------------------------------------------------
__global__ void k_init(float* __restrict__ ws, const float* __restrict__ alpha,
                       float* __restrict__ out) {
  int i = blockIdx.x * 256 + threadIdx.x;
  if (i < WS_RS + B_ * H_ * M_) ws[i] = 0.f;
  if (blockIdx.x == 0) {
    if (threadIdx.x < H_) out[ALPHA_OFF + threadIdx.x] = alpha[threadIdx.x];
    if (threadIdx.x < H_ * M_) out[ZEROS_OFF + threadIdx.x] = 0.f;
  }
}

// ---------------------------------------------------------------------------
// Wt[h][c][k] = wkv_w[k][j(h,c)]  (head-sliced transpose, 2 MB, L2-resident)
// c<64 -> xk column h*64+c ; c>=64 -> xv column 512+h*64+(c-64)
// ---------------------------------------------------------------------------
__global__ void k_build_wt(const float* __restrict__ wkv_w, float* __restrict__ ws) {
  int i = blockIdx.x * 256 + threadIdx.x;
  if (i >= H_ * 128 * HID_) return;
  int h = i / (128 * HID_);
  int r = i % (128 * HID_);
  int c = r / HID_;
  int k = r % HID_;
  int j = (c < 64) ? (h * 64 + c) : (512 + h * 64 + (c - 64));
  ws[WS_WT + i] = wkv_w[(size_t)k * 1024 + j];
}

// ---------------------------------------------------------------------------
// Fused: per (b, h, 64-row n-chunk):
//   kv chunk GEMM (WMMA) -> LDS, scores (WMMA) + softmax over M,
//   slice_weights out, partial slice_token (WMMA) + rowsum atomics
// ---------------------------------------------------------------------------
__global__ __launch_bounds__(256)
void k_fused_slice(const float* __restrict__ x, const float* __restrict__ wkv_b,
                   const float* __restrict__ wtq, const float* __restrict__ alpha,
                   float* __restrict__ out, float* __restrict__ ws) {
  __shared__ float xk_s[64 * LP];    // [n][d]  (padded)
  __shared__ float xvT_s[64 * LP];   // [d][n]  (padded, transposed xv)
  __shared__ float w_s[M_ * 64];     // softmaxed slice weights [m][n]
  __shared__ float st_s[M_ * D_];    // partial slice_token accum

  const int tid  = threadIdx.x;
  const int lane = tid & 31;
  const int wid  = tid >> 5;
  const int l15  = lane & 15;
  const int half = lane >> 4;
  const int kk   = half * 2;

  const int nbase = blockIdx.x * 64;
  const int h     = blockIdx.y;
  const int b     = blockIdx.z;

  const float* Wt = ws + WS_WT + (size_t)h * 128 * HID_;
  const v8f vz = {0.f, 0.f, 0.f, 0.f, 0.f, 0.f, 0.f, 0.f};

  for (int i = tid; i < M_ * D_; i += 256) st_s[i] = 0.f;

  // ---- stage 1: kv chunk = x[b, nbase:+64, :] @ wkv head cols (+bias) ----
  {
    const int wr = wid & 3;   // row tile 0..3 (16 rows each)
    const int wc = wid >> 2;  // col half 0..1 (64 cols each)
    const int r0 = wr * 16;
    const int c0 = wc * 64;
    v8f acc[4] = {vz, vz, vz, vz};
    const size_t xrow = ((size_t)b * N_ + nbase + r0 + l15) * HID_;
    const float* wt0 = Wt + (size_t)(c0 + l15) * HID_;
    for (int k0 = 0; k0 < HID_; k0 += 4) {
      v2f a = *(const v2f*)(x + xrow + k0 + kk);
#pragma unroll
      for (int t = 0; t < 4; ++t) {
        v2f bf = *(const v2f*)(wt0 + (size_t)t * 16 * HID_ + k0 + kk);
        acc[t] = wmma4(a, bf, acc[t]);
      }
    }
    // writeback: side is uniform per wave (wc), keep branch at wave level so
    // the compiler emits straight-line ds_store sequences (no per-store exec
    // mask toggling)
    if (wc == 0) {
#pragma unroll
      for (int t = 0; t < 4; ++t) {
        int   dcol = t * 16 + l15;                 // 0..63 -> xk column (d)
        float bias = wkv_b[h * 64 + dcol];
#pragma unroll
        for (int i = 0; i < 8; ++i) {
          int row = r0 + i + half * 8;
          xk_s[row * LP + dcol] = acc[t][i] + bias;   // [n][d]
        }
      }
    } else {
#pragma unroll
      for (int t = 0; t < 4; ++t) {
        int   dcol = t * 16 + l15;                 // 0..63 -> xv column (d)
        float bias = wkv_b[512 + h * 64 + dcol];
#pragma unroll
        for (int i = 0; i < 8; ++i) {
          int row = r0 + i + half * 8;
          xvT_s[dcol * LP + row] = acc[t][i] + bias;  // [d][n] transposed
        }
      }
    }
  }
  __syncthreads();

  // ---- stage 2: scores = wtq[h] @ xk^T, alpha-scale, softmax over M ----
  const float alphah = alpha[h];
  if (wid < 4) {
    const int nt   = wid;
    const int ncol = nt * 16 + l15;
    v8f s0 = vz, s1 = vz;
    for (int k0 = 0; k0 < 64; k0 += 4) {
      v2f bf = *(const v2f*)&xk_s[ncol * LP + k0 + kk];
      v2f a0 = *(const v2f*)(wtq + (size_t)(h * 32 + l15) * 64 + k0 + kk);
      v2f a1 = *(const v2f*)(wtq + (size_t)(h * 32 + 16 + l15) * 64 + k0 + kk);
      s0 = wmma4(a0, bf, s0);
      s1 = wmma4(a1, bf, s1);
    }
    float e[16];
#pragma unroll
    for (int i = 0; i < 8; ++i) { e[i] = s0[i] * alphah; e[8 + i] = s1[i] * alphah; }
    float vmax = -1e30f;
#pragma unroll
    for (int i = 0; i < 16; ++i) vmax = fmaxf(vmax, e[i]);
    vmax = fmaxf(vmax, __shfl_xor(vmax, 16, 32));
    float sum = 0.f;
#pragma unroll
    for (int i = 0; i < 16; ++i) { e[i] = __expf(e[i] - vmax); sum += e[i]; }
    sum += __shfl_xor(sum, 16, 32);
    float inv = 1.f / sum;
    size_t swb = (size_t)SW_OFF + ((size_t)b * H_ + h) * M_ * N_ + nbase + ncol;
#pragma unroll
    for (int i = 0; i < 8; ++i) {
      int   m0 = i + half * 8;
      int   m1 = 16 + i + half * 8;
      float w0 = e[i] * inv, w1 = e[8 + i] * inv;
      w_s[m0 * 64 + ncol] = w0;
      w_s[m1 * 64 + ncol] = w1;
      out[swb + (size_t)m0 * N_] = w0;
      out[swb + (size_t)m1 * N_] = w1;
    }
  }
  __syncthreads();

  // ---- stage 3: partial slice_token = W(32x64n) @ xv(64n x 64d) ----
  {
    v8f acc[2][4] = {{vz, vz, vz, vz}, {vz, vz, vz, vz}};
    const int kb = wid * 8;  // K(n) slice per wave
    for (int k0 = kb; k0 < kb + 8; k0 += 4) {
      v2f a0 = *(const v2f*)&w_s[l15 * 64 + k0 + kk];
      v2f a1 = *(const v2f*)&w_s[(16 + l15) * 64 + k0 + kk];
#pragma unroll
      for (int dt = 0; dt < 4; ++dt) {
        int dcol = dt * 16 + l15;
        // B[k=n][c=d] = xv[n][d] = xvT_s[d][n] -> contiguous b64 pair
        v2f bf = *(const v2f*)&xvT_s[dcol * LP + k0 + kk];
        acc[0][dt] = wmma4(a0, bf, acc[0][dt]);
        acc[1][dt] = wmma4(a1, bf, acc[1][dt]);
      }
    }
#pragma unroll
    for (int mt = 0; mt < 2; ++mt)
#pragma unroll
      for (int dt = 0; dt < 4; ++dt)
#pragma unroll
        for (int i = 0; i < 8; ++i) {
          int m = mt * 16 + i + half * 8;
          int d = dt * 16 + l15;
          atomicAdd(&st_s[m * 64 + d], acc[mt][dt][i]);
        }
  }
  __syncthreads();

  for (int i = tid; i < M_ * D_; i += 256) {
    int m = i >> 6, d = i & 63;
    atomicAdd(&ws[WS_ST + (((size_t)b * H_ + h) * M_ + m) * D_ + d], st_s[i]);
  }
  if (tid < M_) {
    float rs = 0.f;
    for (int n = 0; n < 64; ++n) rs += w_s[tid * 64 + n];
    atomicAdd(&ws[WS_RS + ((size_t)b * H_ + h) * M_ + tid], rs);
  }
}

// ---------------------------------------------------------------------------
// normalize slice_token -> (B, M, HID) layout
// ---------------------------------------------------------------------------
__global__ void k_norm_st(float* __restrict__ ws) {
  int i = blockIdx.x * 256 + threadIdx.x;
  if (i >= B_ * M_ * HID_) return;
  int b = i / (M_ * HID_);
  int r = i % (M_ * HID_);
  int m = r / HID_;
  int j = r % HID_;
  int h = j / D_, d = j % D_;
  float rs = ws[WS_RS + ((size_t)b * H_ + h) * M_ + m];
  float v  = ws[WS_ST + (((size_t)b * H_ + h) * M_ + m) * D_ + d];
  ws[WS_STN + i] = v / (rs + 1e-5f);
}

// ---------------------------------------------------------------------------
// tiny per-(b,h): qkv projection, 32-token MHA, attn output,
// fold out_w through out_token, store transposed: Gt[b][c][h*32+m]
// ---------------------------------------------------------------------------
__global__ __launch_bounds__(256)
void k_small_attn(const float* __restrict__ qkv_w, const float* __restrict__ out_w,
                  float* __restrict__ ws, float* __restrict__ out) {
  __shared__ float q_s[M_ * D_], k_s[M_ * D_], v_s[M_ * D_];
  __shared__ float at_s[M_ * M_], ot_s[M_ * D_];
  const int tid = threadIdx.x;
  const int h = blockIdx.x, b = blockIdx.y;
  const float* stn = ws + WS_STN + (size_t)b * M_ * HID_;

  for (int idx = tid; idx < 3 * M_ * D_; idx += 256) {
    int which = idx / (M_ * D_);
    int r = idx % (M_ * D_);
    int m = r / D_, d = r % D_;
    int col = which * HID_ + h * D_ + d;
    float s = 0.f;
    for (int j = 0; j < HID_; ++j)
      s += stn[m * HID_ + j] * qkv_w[(size_t)j * (3 * HID_) + col];
    float* dst = (which == 0) ? q_s : (which == 1) ? k_s : v_s;
    dst[m * D_ + d] = s;
  }
  __syncthreads();

  for (int idx = tid; idx < M_ * M_; idx += 256) {
    int m = idx / M_, n2 = idx % M_;
    float s = 0.f;
    for (int d = 0; d < D_; ++d) s += q_s[m * D_ + d] * k_s[n2 * D_ + d];
    at_s[idx] = s * 0.125f;  // D^-0.5
  }
  __syncthreads();

  if (tid < M_) {
    int m = tid;
    float mx = -1e30f;
    for (int k = 0; k < M_; ++k) mx = fmaxf(mx, at_s[m * M_ + k]);
    float sum = 0.f;
    for (int k = 0; k < M_; ++k) {
      float e = __expf(at_s[m * M_ + k] - mx);
      at_s[m * M_ + k] = e;
      sum += e;
    }
    float inv = 1.f / sum;
    for (int k = 0; k < M_; ++k) {
      float w = at_s[m * M_ + k] * inv;
      at_s[m * M_ + k] = w;
      out[(size_t)ATTN_OFF + (((size_t)b * H_ + h) * M_ + m) * M_ + k] = w;
    }
  }
  __syncthreads();

  for (int idx = tid; idx < M_ * D_; idx += 256) {
    int m = idx / D_, d = idx % D_;
    float s = 0.f;
    for (int k = 0; k < M_; ++k) s += at_s[m * M_ + k] * v_s[k * D_ + d];
    ot_s[idx] = s;
  }
  __syncthreads();

  for (int idx = tid; idx < M_ * HID_; idx += 256) {
    int m = idx / HID_, c = idx % HID_;
    float s = 0.f;
    for (int d = 0; d < D_; ++d)
      s += ot_s[m * D_ + d] * out_w[(size_t)(h * D_ + d) * HID_ + c];
    // transposed: Gt[b][c][h*32+m]
    ws[WS_GT + ((size_t)b * HID_ + c) * (H_ * M_) + h * M_ + m] = s;
  }
}

// ---------------------------------------------------------------------------
// final: out[b] = SW_cat[b] (N x 256) @ G[b] (256 x 512) + out_b   (WMMA)
// B-fragments from transposed Gt -> single b64 loads
// ---------------------------------------------------------------------------
__global__ __launch_bounds__(256)
void k_out_gemm(const float* __restrict__ ws, const float* __restrict__ out_b,
                float* out) {
  const int tid  = threadIdx.x;
  const int lane = tid & 31, wid = tid >> 5;
  const int l15  = lane & 15, half = lane >> 4, kk = half * 2;
  const int nbase = blockIdx.x * 128;
  const int cbase = blockIdx.y * 128;
  const int b     = blockIdx.z;

  const float* sw = out + SW_OFF;   // A[n][k] = sw[(b*256+k)*N + n]
  const float* Gt = ws + WS_GT + (size_t)b * HID_ * (H_ * M_);

  const v8f vz = {0.f, 0.f, 0.f, 0.f, 0.f, 0.f, 0.f, 0.f};
  v8f acc[8] = {vz, vz, vz, vz, vz, vz, vz, vz};

  const int nrow = nbase + wid * 16 + l15;
  for (int k0 = 0; k0 < H_ * M_; k0 += 4) {
    v2f a;
    a.x = sw[((size_t)b * 256 + k0 + kk) * N_ + nrow];
    a.y = sw[((size_t)b * 256 + k0 + kk + 1) * N_ + nrow];
#pragma unroll
    for (int t = 0; t < 8; ++t) {
      int c = cbase + t * 16 + l15;
      v2f bf = *(const v2f*)(Gt + (size_t)c * (H_ * M_) + k0 + kk);
      acc[t] = wmma4(a, bf, acc[t]);
    }
  }
#pragma unroll
  for (int t = 0; t < 8; ++t) {
    int   c    = cbase + t * 16 + l15;
    float bias = out_b[c];
#pragma unroll
    for (int i = 0; i < 8; ++i) {
      int n = nbase + wid * 16 + i + half * 8;
      out[((size_t)b * N_ + n) * HID_ + c] = acc[t][i] + bias;
    }
  }
}

// ---------------------------------------------------------------------------
extern "C" void kernel_launch(void* const* d_in, const int* in_sizes, int n_in,
                              void* d_out, int out_size, void* d_ws, size_t ws_size,
                              hipStream_t stream) {
  const float* x      = (const float*)d_in[0];
  const float* wkv_w  = (const float*)d_in[1];
  const float* wkv_b  = (const float*)d_in[2];
  const float* wtq    = (const float*)d_in[3];
  const float* alpha  = (const float*)d_in[4];
  const float* qkv_w  = (const float*)d_in[5];
  const float* out_w  = (const float*)d_in[6];
  const float* out_b  = (const float*)d_in[7];
  float* out = (float*)d_out;
  float* ws  = (float*)d_ws;

  k_init<<<(WS_RS + B_ * H_ * M_ + 255) / 256, 256, 0, stream>>>(ws, alpha, out);
  k_build_wt<<<(H_ * 128 * HID_ + 255) / 256, 256, 0, stream>>>(wkv_w, ws);
  k_fused_slice<<<dim3(N_ / 64, H_, B_), 256, 0, stream>>>(x, wkv_b, wtq, alpha,
                                                           out, ws);
  k_norm_st<<<(B_ * M_ * HID_ + 255) / 256, 256, 0, stream>>>(ws);
  k_small_attn<<<dim3(H_, B_), 256, 0, stream>>>(qkv_w, out_w, ws, out);
  k_out_gemm<<<dim3(N_ / 128, HID_ / 128, B_), 256, 0, stream>>>(ws, out_b, out);
}